// LoadBalanceLoss_18081812316843
// MI455X (gfx1250) — compile-verified
//
#include <hip/hip_runtime.h>
#include <hip/hip_bf16.h>

typedef __attribute__((ext_vector_type(2))) float     v2f;
typedef __attribute__((ext_vector_type(8))) float     v8f;
typedef __attribute__((ext_vector_type(2))) long long v2ll;

#define NUM_EXPERTS 64

// Grid for the column-sum kernel is a compile-time constant so the unrolled
// tile strides fold into 24-bit signed immediate offsets (2MB/4MB/6MB < 8.38MB).
#define COLSUM_GRID    1024
#define COLSUM_STREAMS (COLSUM_GRID * 2)            // 2 row-halves per block
#define TILE_STEP_F    ((long long)COLSUM_STREAMS * 4 * NUM_EXPERTS) // floats between a stream's tiles

#define HIST_GRID 512

// ---------------------------------------------------------------------------
// Zero the workspace accumulators (P_sum[64] f32 + counts[64] u32).
// Must run every call: harness poisons d_ws with 0xAA and never re-zeros.
// ---------------------------------------------------------------------------
__global__ void lbl_zero_ws(unsigned int* ws) {
    int i = threadIdx.x;
    if (i < 2 * NUM_EXPERTS) ws[i] = 0u;
}

// ---------------------------------------------------------------------------
// Histogram of expert_indices (int64, values in [0,64)) into counts[64].
// 16B vector loads (global_load_b128), LDS ds_add_u32 bins, then 64 global
// atomics per workgroup. Non-temporal: indices are streamed exactly once.
// ---------------------------------------------------------------------------
__global__ __launch_bounds__(256) void lbl_hist(const long long* __restrict__ idx,
                                                unsigned int* __restrict__ counts,
                                                long long n) {
    __shared__ unsigned int bins[NUM_EXPERTS];
    const int tid = threadIdx.x;
    if (tid < NUM_EXPERTS) bins[tid] = 0u;
    __syncthreads();

    const v2ll* __restrict__ idx2 = (const v2ll*)idx;
    const long long npair  = n >> 1;
    const long long stride = (long long)HIST_GRID * 256;
    for (long long i = (long long)blockIdx.x * 256 + tid; i < npair; i += stride) {
        const v2ll v = __builtin_nontemporal_load(&idx2[i]);
        atomicAdd(&bins[(int)v.x & (NUM_EXPERTS - 1)], 1u);
        atomicAdd(&bins[(int)v.y & (NUM_EXPERTS - 1)], 1u);
    }
    // odd-length tail (not hit for B*topk even, kept for generality)
    if ((n & 1) && blockIdx.x == 0 && tid == 0) {
        atomicAdd(&bins[(int)idx[n - 1] & (NUM_EXPERTS - 1)], 1u);
    }
    __syncthreads();
    if (tid < NUM_EXPERTS) {
        const unsigned int v = bins[tid];
        if (v) atomicAdd(&counts[tid], v);
    }
}

// ---------------------------------------------------------------------------
// Column sums of router_probs (B x 64, f32) via V_WMMA_F32_16X16X4_F32.
// D = ones(16x4) * Btile(4x16) + C : every output row = column partial sums.
//
// B-tile 4x16 operand layout (wave32):
//   lane l<16  -> column l,    VGPR0 = row K=0, VGPR1 = row K=1
//   lane l>=16 -> column l-16, VGPR0 = row K=2, VGPR1 = row K=3
// C/D: VGPR0 lanes 0..15 = (M=0, N=lane) -> column sums read from there.
//
// 8 waves/block: wave&3 = 16-expert column group, wave>>2 = row-half.
// Unroll 4: 8 immediate-offset NT loads clause together, then 4 chained WMMAs
// drain them with staged loadcnt waits (4x the in-flight loads per wave).
// EXEC is all-ones for every WMMA (uniform scalar trip count, no divergence).
// ---------------------------------------------------------------------------
__global__ __launch_bounds__(256) void lbl_colsum_wmma(const float* __restrict__ probs,
                                                       float* __restrict__ psum,
                                                       long long B) {
    __shared__ float lds[NUM_EXPERTS];
    const int tid = threadIdx.x;
    if (tid < NUM_EXPERTS) lds[tid] = 0.0f;
    __syncthreads();

    const int lane     = tid & 31;
    const int wave     = tid >> 5;
    const int colgroup = wave & 3;   // which 16-expert slice
    const int rowhalf  = wave >> 2;  // 0 or 1

    const int col  = colgroup * 16 + (lane & 15);
    const int krow = (lane >> 4) * 2;          // rows {0,1} or {2,3} of a tile

    v2f a; a.x = 1.0f; a.y = 1.0f;             // ones matrix (reduction weights)
    v8f c = {};                                // f32 accumulator

    const long long ntiles = B >> 2;           // 4 rows per WMMA
    const long long t0     = (long long)blockIdx.x * 2 + rowhalf;
    const long long nt     = (t0 < ntiles) ? ((ntiles - 1 - t0) / COLSUM_STREAMS + 1) : 0;

    const float* p = probs + (t0 * 4 + krow) * NUM_EXPERTS + col;

    long long i = 0;
    for (; i + 4 <= nt; i += 4) {
        v2f b0, b1, b2, b3;
        b0.x = __builtin_nontemporal_load(p);
        b0.y = __builtin_nontemporal_load(p + NUM_EXPERTS);
        b1.x = __builtin_nontemporal_load(p + TILE_STEP_F);
        b1.y = __builtin_nontemporal_load(p + TILE_STEP_F + NUM_EXPERTS);
        b2.x = __builtin_nontemporal_load(p + 2 * TILE_STEP_F);
        b2.y = __builtin_nontemporal_load(p + 2 * TILE_STEP_F + NUM_EXPERTS);
        b3.x = __builtin_nontemporal_load(p + 3 * TILE_STEP_F);
        b3.y = __builtin_nontemporal_load(p + 3 * TILE_STEP_F + NUM_EXPERTS);
        c = __builtin_amdgcn_wmma_f32_16x16x4_f32(false, a, false, b0, (short)0, c, false, false);
        c = __builtin_amdgcn_wmma_f32_16x16x4_f32(false, a, false, b1, (short)0, c, false, false);
        c = __builtin_amdgcn_wmma_f32_16x16x4_f32(false, a, false, b2, (short)0, c, false, false);
        c = __builtin_amdgcn_wmma_f32_16x16x4_f32(false, a, false, b3, (short)0, c, false, false);
        p += 4 * TILE_STEP_F;
    }
    for (; i < nt; ++i) {
        v2f b;
        b.x = __builtin_nontemporal_load(p);
        b.y = __builtin_nontemporal_load(p + NUM_EXPERTS);
        c = __builtin_amdgcn_wmma_f32_16x16x4_f32(false, a, false, b, (short)0, c, false, false);
        p += TILE_STEP_F;
    }

    // Lanes 0..15 of c[0] hold the column sums (row M=0 of D; all rows equal).
    if (lane < 16) atomicAdd(&lds[colgroup * 16 + lane], c[0]);
    __syncthreads();
    if (tid < NUM_EXPERTS) atomicAdd(&psum[tid], lds[tid]);
}

// ---------------------------------------------------------------------------
// loss = num_experts * sum_m (counts[m]/(B*topk)) * (Psum[m]/B)
//      = scale * sum_m counts[m]*Psum[m],  scale = num_experts/(B*topk*B)
// ---------------------------------------------------------------------------
__global__ void lbl_finalize(const float* __restrict__ psum,
                             const unsigned int* __restrict__ counts,
                             float* __restrict__ out, float scale) {
    __shared__ float red[NUM_EXPERTS];
    const int tid = threadIdx.x;               // 64 threads
    red[tid] = psum[tid] * (float)counts[tid];
    __syncthreads();
    for (int s = NUM_EXPERTS / 2; s > 0; s >>= 1) {
        if (tid < s) red[tid] += red[tid + s];
        __syncthreads();
    }
    if (tid == 0) out[0] = red[0] * scale;
}

extern "C" void kernel_launch(void* const* d_in, const int* in_sizes, int n_in,
                              void* d_out, int out_size, void* d_ws, size_t ws_size,
                              hipStream_t stream) {
    const float*     probs = (const float*)d_in[0];
    const long long* idx   = (const long long*)d_in[1];

    const long long nprobs = (long long)in_sizes[0];          // B * 64
    const long long nidx   = (long long)in_sizes[1];          // B * topk
    const long long B      = nprobs / NUM_EXPERTS;
    const double    scaled = (double)NUM_EXPERTS / ((double)nidx * (double)B);

    float*        psum   = (float*)d_ws;                      // 64 f32
    unsigned int* counts = (unsigned int*)d_ws + NUM_EXPERTS; // 64 u32
    float*        out    = (float*)d_out;

    lbl_zero_ws<<<1, 2 * NUM_EXPERTS, 0, stream>>>((unsigned int*)d_ws);

    lbl_hist<<<HIST_GRID, 256, 0, stream>>>(idx, counts, nidx);

    lbl_colsum_wmma<<<COLSUM_GRID, 256, 0, stream>>>(probs, psum, B);

    lbl_finalize<<<1, NUM_EXPERTS, 0, stream>>>(psum, counts, out, (float)scaled);
}